// MLP_4071628997356
// MI455X (gfx1250) — compile-verified
//
#include <hip/hip_runtime.h>

typedef __attribute__((ext_vector_type(2))) float v2f;
typedef __attribute__((ext_vector_type(2), aligned(4))) float v2f_u4; // 4B-aligned loads (rows are 4097 floats)
typedef __attribute__((ext_vector_type(4))) float v4f;
typedef __attribute__((ext_vector_type(8))) float v8f;

#define X_DIM 4096
#define Z_DIM 64
#define N_HIDDEN 1024
#define ROW_LEN (X_DIM + 1)              // 4097
#define ROWS (Z_DIM * N_HIDDEN)          // 65536
#define TILE_ROWS 16
#define TILES (ROWS / TILE_ROWS)         // 4096
#define WAVES_PER_BLOCK 8
#define TILES_PER_HEAD (N_HIDDEN / TILE_ROWS) // 64

// One wave per 16-row tile of W1. D = A(16x4 tile of W1 rows) * B(4x16 of x
// broadcast) accumulated over K with v_wmma_f32_16x16x4_f32.
__global__ __launch_bounds__(256) void mlp_gemv_wmma(
    const float* __restrict__ x, const float* __restrict__ z,
    const float* __restrict__ W1, const float* __restrict__ b1,
    const float* __restrict__ W2, float* __restrict__ ws)
{
    __shared__ __align__(16) float xs[X_DIM];
    __shared__ float rowdot[WAVES_PER_BLOCK][TILE_ROWS];

    // Stage x into LDS (coalesced b128 loads), 16 KB per workgroup.
    {
        const v4f* xi = (const v4f*)x;
        v4f* xo = (v4f*)xs;
        for (int i = threadIdx.x; i < X_DIM / 4; i += 256) xo[i] = xi[i];
    }
    __syncthreads();

    const int lane = threadIdx.x & 31;
    const int wv   = threadIdx.x >> 5;
    const int tile = blockIdx.x * WAVES_PER_BLOCK + wv;
    const int half = lane >> 4;   // 0: lanes 0-15 (K=0,1), 1: lanes 16-31 (K=2,3)
    const int m    = lane & 15;   // row-in-tile for A operand

    // A operand stream: lane L reads row (tile*16 + m), columns k0 + 2*half + {0,1}
    const float* ap = W1 + (size_t)(tile * TILE_ROWS + m) * ROW_LEN + 2 * half;
    const int    boff = 2 * half; // B operand: x[k0 + 2*half + {0,1}] from LDS

    v8f acc = {0.f, 0.f, 0.f, 0.f, 0.f, 0.f, 0.f, 0.f};

    #pragma unroll 8
    for (int k0 = 0; k0 < X_DIM; k0 += 4) {
        v2f a = (v2f)(*(const v2f_u4*)(ap + k0));   // W1 tile slice (may be 4B aligned)
        v2f b;
        b.x = xs[k0 + boff];
        b.y = xs[k0 + boff + 1];
        acc = __builtin_amdgcn_wmma_f32_16x16x4_f32(
            /*neg_a=*/false, a, /*neg_b=*/false, b,
            /*c_mod=*/(short)0, acc, /*reuse_a=*/false, /*reuse_b=*/false);
    }

    // D layout: VGPR r, lanes 0-15 -> M=r (col=lane), lanes 16-31 -> M=8+r.
    // All columns are equal (B columns identical), so take column 0 / 16.
    if (lane == 0) {
        #pragma unroll
        for (int j = 0; j < 8; ++j) rowdot[wv][j] = acc[j];
    } else if (lane == 16) {
        #pragma unroll
        for (int j = 0; j < 8; ++j) rowdot[wv][8 + j] = acc[j];
    }
    __syncthreads();

    float c = 0.f;
    if (lane < 16) {
        const int row  = tile * TILE_ROWS + m;
        const int head = tile / TILES_PER_HEAD;
        float dotv  = rowdot[wv][m];
        float wlast = W1[(size_t)row * ROW_LEN + X_DIM]; // column 4096 multiplies z_k
        float pre   = dotv + wlast * z[head] + b1[row];
        float h     = pre > 0.f ? pre : 0.f;
        c = h * W2[row];
    }
    // Reduce the 16 per-row contributions (lanes 0-15) down to lane 0.
    c += __shfl_xor(c, 1, 32);
    c += __shfl_xor(c, 2, 32);
    c += __shfl_xor(c, 4, 32);
    c += __shfl_xor(c, 8, 32);
    if (lane == 0) ws[tile] = c;
}

// Deterministic per-head reduction of 64 tile partials + b2 (no float atomics).
__global__ __launch_bounds__(64) void mlp_reduce(
    const float* __restrict__ ws, const float* __restrict__ b2,
    float* __restrict__ out)
{
    int k = threadIdx.x;
    if (k < Z_DIM) {
        float s = b2[k];
        #pragma unroll 8
        for (int t = 0; t < TILES_PER_HEAD; ++t) s += ws[k * TILES_PER_HEAD + t];
        out[k] = s;
    }
}

extern "C" void kernel_launch(void* const* d_in, const int* in_sizes, int n_in,
                              void* d_out, int out_size, void* d_ws, size_t ws_size,
                              hipStream_t stream) {
    const float* x  = (const float*)d_in[0];
    const float* z  = (const float*)d_in[1];
    const float* W1 = (const float*)d_in[2];
    const float* b1 = (const float*)d_in[3];
    const float* W2 = (const float*)d_in[4];
    const float* b2 = (const float*)d_in[5];
    float* out = (float*)d_out;
    float* ws  = (float*)d_ws;   // TILES floats of per-tile partial sums

    mlp_gemv_wmma<<<TILES / WAVES_PER_BLOCK, 256, 0, stream>>>(x, z, W1, b1, W2, ws);
    mlp_reduce<<<1, 64, 0, stream>>>(ws, b2, out);
}